// VGP_62466004353508
// MI455X (gfx1250) — compile-verified
//
// GP-VAE forward for MI455X (gfx1250), FP32 WMMA everywhere.
// All n^3 work (Cholesky SYRK, blocked TRSM updates, kk_inv GEMMs, linear
// layers) runs on V_WMMA_F32_16X16X4_F32. Working set is L2-resident
// (~50MB << 192MB global L2), so GEMMs load straight from L2 per k-step.
// GEMM inner loop is branch-free, all loads forced to addrspace(1)
// (global_load_*, LOADcnt-only waits), and ping-pong double-buffered:
// slot A fragments are consumed by WMMAs while slot B loads are in flight,
// with zero register-rotation movs.
//
// Input flattening assumption (setup_inputs dict order, nested params dict
// in insertion order, each fc = {W, b}):
//  0:x  1:fc1.W 2:fc1.b 3:fc11.W 4:fc11.b 5:fc12.W 6:fc12.b 7:fc2.W 8:fc2.b
//  9:fc21.W 10:fc21.b 11:fc22.W 12:fc22.b 13:fc3.W 14:fc3.b 15:fc31.W
// 16:fc31.b 17:fc32.W 18:fc32.b 19:fc4.W 20:fc4.b 21:fc41.W 22:fc41.b
// 23:fc42.W 24:fc42.b 25:s 26:t 27:sigma2 28:w 29:eps_xi 30:eps_f 31:eps_z
//
// d_out: [kld_loss, nll_loss, z_mean(2048x1088), z_covh(2048x36),
//         x_mean(2048x128), x_lcov(2048x128)]

#include <hip/hip_runtime.h>
#include <math.h>

typedef float v2f __attribute__((ext_vector_type(2)));
typedef float v8f __attribute__((ext_vector_type(8)));

#define GF_TRANSA 1
#define GF_TRANSB 2
#define GF_RELU   4
#define GF_NEG    8

#define NB      2048
#define NBLK    32          // 2048 / 64
#define NPB     256         // partial blocks per reduction
#define LOG2PI_F 1.8378770664093453f

// force global (addrspace 1) loads: flat_load would tick DScnt too and insert
// s_wait_loadcnt_dscnt 0x0 before every WMMA.
__device__ __forceinline__ v2f ldg_pair(const float* p, size_t stride2) {
  const __attribute__((address_space(1))) float* g =
      (const __attribute__((address_space(1))) float*)p;
  v2f r;
  r.x = g[0];
  r.y = g[stride2];
  return r;
}

// ---------------------------------------------------------------- WMMA GEMM
// C(MxN) = [Cin + bias] (+/-) op(A) @ op(B); op per flags. Requires K%4==0.
// Wave tile: 32(M) x 64(N) -> 8 independent v_wmma_f32_16x16x4_f32 chains.
// Workgroup: 4 waves stacked in M -> 128x64 per block.
// Fragment layouts (f32):
//   A 16x4 : lane%16 = M row, VGPR v -> K = kk + v + 2*(lane/16)
//   B 4x16 : lane%16 = N col, VGPR v -> K = kk + v + 2*(lane/16)
//   C 16x16: lane%16 = N col, VGPR v -> M = v (lanes 0-15) / v+8 (lanes 16-31)
template <int FLAGS>
__global__ __launch_bounds__(128) void gemm_wmma_f32(
    float* C, int ldc, const float* Cin, int ldcin,
    const float* __restrict__ A, int lda,
    const float* __restrict__ B, int ldb,
    const float* __restrict__ bias, int M, int N, int K) {
  const int lane = threadIdx.x & 31;
  const int wave = threadIdx.x >> 5;
  const int l16  = lane & 15;
  const int half = lane >> 4;
  const int m0   = blockIdx.y * 128 + wave * 32;
  const int n0   = blockIdx.x * 64;

  // clamped row/col indices: out-of-range fragments load valid garbage that
  // the guarded epilogue never stores (no predication in the hot loop).
  int amr[2];
#pragma unroll
  for (int t = 0; t < 2; ++t) {
    int m = m0 + t * 16 + l16;
    amr[t] = m < M ? m : M - 1;
  }
  int ncol[4];
#pragma unroll
  for (int j = 0; j < 4; ++j) {
    int n = n0 + j * 16 + l16;
    ncol[j] = n < N ? n : N - 1;
  }

  // pointer setup (trans selects fold at compile time via template FLAGS)
  const float* ap[2];
  size_t astep, a2;
  if (!(FLAGS & GF_TRANSA)) {
    ap[0] = A + (size_t)amr[0] * lda + 2 * half;
    ap[1] = A + (size_t)amr[1] * lda + 2 * half;
    astep = 4; a2 = 1;
  } else {
    ap[0] = A + (size_t)(2 * half) * lda + amr[0];
    ap[1] = A + (size_t)(2 * half) * lda + amr[1];
    astep = 4 * (size_t)lda; a2 = (size_t)lda;
  }
  const float* bp[4];
  size_t bstep, b2;
  if (!(FLAGS & GF_TRANSB)) {
#pragma unroll
    for (int j = 0; j < 4; ++j) bp[j] = B + (size_t)(2 * half) * ldb + ncol[j];
    bstep = 4 * (size_t)ldb; b2 = (size_t)ldb;
  } else {
#pragma unroll
    for (int j = 0; j < 4; ++j) bp[j] = B + (size_t)ncol[j] * ldb + 2 * half;
    bstep = 4; b2 = 1;
  }

  v8f acc[2][4];
#pragma unroll
  for (int t = 0; t < 2; ++t)
#pragma unroll
    for (int j = 0; j < 4; ++j)
#pragma unroll
      for (int v = 0; v < 8; ++v) acc[t][j][v] = 0.0f;

  // ping-pong software pipeline: while the WMMA block consumes slot s, the
  // loads for slot 1-s are in flight. Compile-time slot indices -> no
  // register-rotation movs in the steady state.
  v2f af[2][2];        // [slot][m-tile]
  v2f bf[2][4];        // [slot][n-tile]
  af[0][0] = ldg_pair(ap[0], a2);
  af[0][1] = ldg_pair(ap[1], a2);
#pragma unroll
  for (int j = 0; j < 4; ++j) bf[0][j] = ldg_pair(bp[j], b2);

  int kk = 4;
  for (;;) {
    // consume slot 0, prefetch slot 1
    const bool more0 = kk < K;
    if (more0) {
      ap[0] += astep; ap[1] += astep;
      af[1][0] = ldg_pair(ap[0], a2);
      af[1][1] = ldg_pair(ap[1], a2);
      __builtin_prefetch(ap[0] + 4 * astep, 0, 3);
#pragma unroll
      for (int j = 0; j < 4; ++j) {
        bp[j] += bstep;
        bf[1][j] = ldg_pair(bp[j], b2);
      }
    }
#pragma unroll
    for (int j = 0; j < 4; ++j) {
      acc[0][j] = __builtin_amdgcn_wmma_f32_16x16x4_f32(
          false, af[0][0], false, bf[0][j], (short)0, acc[0][j], false, false);
      acc[1][j] = __builtin_amdgcn_wmma_f32_16x16x4_f32(
          false, af[0][1], false, bf[0][j], (short)0, acc[1][j], false, false);
    }
    if (!more0) break;
    kk += 4;

    // consume slot 1, prefetch slot 0
    const bool more1 = kk < K;
    if (more1) {
      ap[0] += astep; ap[1] += astep;
      af[0][0] = ldg_pair(ap[0], a2);
      af[0][1] = ldg_pair(ap[1], a2);
      __builtin_prefetch(ap[0] + 4 * astep, 0, 3);
#pragma unroll
      for (int j = 0; j < 4; ++j) {
        bp[j] += bstep;
        bf[0][j] = ldg_pair(bp[j], b2);
      }
    }
#pragma unroll
    for (int j = 0; j < 4; ++j) {
      acc[0][j] = __builtin_amdgcn_wmma_f32_16x16x4_f32(
          false, af[1][0], false, bf[1][j], (short)0, acc[0][j], false, false);
      acc[1][j] = __builtin_amdgcn_wmma_f32_16x16x4_f32(
          false, af[1][1], false, bf[1][j], (short)0, acc[1][j], false, false);
    }
    if (!more1) break;
    kk += 4;
  }

#pragma unroll
  for (int t = 0; t < 2; ++t)
#pragma unroll
    for (int j = 0; j < 4; ++j) {
      const int n = n0 + j * 16 + l16;
      if (n >= N) continue;
#pragma unroll
      for (int v = 0; v < 8; ++v) {
        const int m = m0 + t * 16 + v + 8 * half;
        if (m >= M) continue;
        float init = 0.f;
        if (Cin)  init  = Cin[(size_t)m * ldcin + n];
        if (bias) init += bias[n];
        const float p = acc[t][j][v];
        float out = (FLAGS & GF_NEG) ? (init - p) : (init + p);
        if (FLAGS & GF_RELU) out = out > 0.f ? out : 0.f;
        C[(size_t)m * ldc + n] = out;
      }
    }
}

// ------------------------------------------------------- Cholesky (64 blocks)
__global__ __launch_bounds__(256) void chol_diag(float* Kss, int ld, int k0) {
  __shared__ float S[64][65];
  __shared__ float sinv;
  const int tid = threadIdx.x;
  float* base = Kss + (size_t)k0 * ld + k0;
  for (int idx = tid; idx < 64 * 64; idx += 256)
    S[idx / 64][idx % 64] = base[(size_t)(idx / 64) * ld + (idx % 64)];
  __syncthreads();
  for (int j = 0; j < 64; ++j) {
    if (tid == 0) {
      float d = sqrtf(S[j][j]);
      S[j][j] = d;
      sinv = 1.f / d;
    }
    __syncthreads();
    const float inv = sinv;
    for (int i = j + 1 + tid; i < 64; i += 256) S[i][j] *= inv;
    __syncthreads();
    for (int i = j + 1 + tid; i < 64; i += 256) {
      const float lij = S[i][j];
      for (int c = j + 1; c <= i; ++c) S[i][c] -= lij * S[c][j];
    }
    __syncthreads();
  }
  for (int idx = tid; idx < 64 * 64; idx += 256)
    base[(size_t)(idx / 64) * ld + (idx % 64)] = S[idx / 64][idx % 64];
}

// Panel solve: X = A * inv(L_kk)^T for a 64x64 block below the diagonal.
__global__ __launch_bounds__(64) void chol_trsm_panel(float* Kss, int ld, int k0) {
  __shared__ float Lk[64][65];
  __shared__ float X[64][65];
  const int tid = threadIdx.x;              // row owner
  const float* Ld = Kss + (size_t)k0 * ld + k0;
  const int r0 = k0 + 64 * (1 + blockIdx.x);
  float* Ablk = Kss + (size_t)r0 * ld + k0;
  for (int idx = tid; idx < 64 * 64; idx += 64) {
    const int r = idx / 64, c = idx % 64;
    Lk[r][c] = Ld[(size_t)r * ld + c];
    X[r][c]  = Ablk[(size_t)r * ld + c];
  }
  __syncthreads();
  const int i = tid;                        // each thread solves its own row
  for (int j = 0; j < 64; ++j) {
    const float xij = X[i][j] / Lk[j][j];
    X[i][j] = xij;
    for (int c = j + 1; c < 64; ++c) X[i][c] -= xij * Lk[c][j];
  }
  for (int c = 0; c < 64; ++c) Ablk[(size_t)i * ld + c] = X[i][c];
}

// 64-deep triangular solve, one thread per RHS column, fully register-resident.
__global__ __launch_bounds__(256) void trsv_slab(const float* L, int ldL,
                                                 float* Bm, int ldb, int transposed) {
  __shared__ float Ls[64][65];
  __shared__ float invd[64];
  const int tid = threadIdx.x;
  for (int idx = tid; idx < 64 * 64; idx += 256)
    Ls[idx / 64][idx % 64] = L[(size_t)(idx / 64) * ldL + (idx % 64)];
  __syncthreads();
  if (tid < 64) invd[tid] = 1.f / Ls[tid][tid];
  __syncthreads();
  const int c = blockIdx.x * 256 + tid;
  float w[64];
#pragma unroll
  for (int r = 0; r < 64; ++r) w[r] = Bm[(size_t)r * ldb + c];
  if (!transposed) {
#pragma unroll
    for (int r = 0; r < 64; ++r) {
      const float t = w[r] * invd[r];
      w[r] = t;
#pragma unroll
      for (int rr = r + 1; rr < 64; ++rr) w[rr] -= t * Ls[rr][r];
    }
  } else {
#pragma unroll
    for (int r = 63; r >= 0; --r) {
      const float t = w[r] * invd[r];
      w[r] = t;
#pragma unroll
      for (int rr = 0; rr < r; ++rr) w[rr] -= t * Ls[r][rr];
    }
  }
#pragma unroll
  for (int r = 0; r < 64; ++r) Bm[(size_t)r * ldb + c] = w[r];
}

// --------------------------------------------------------------- elementwise
__global__ void rbf_kernel(float* out, const float* __restrict__ X,
                           const float* __restrict__ Y,
                           const float* __restrict__ w,
                           const float* __restrict__ sigma2_p,
                           int n, int f, float jitter) {
  const size_t idx = (size_t)blockIdx.x * blockDim.x + threadIdx.x;
  if (idx >= (size_t)n * n) return;
  const int j = (int)(idx / n);   // Y row
  const int i = (int)(idx % n);   // X row (contiguous across lanes)
  const float* xr = X + (size_t)i * f;
  const float* yr = Y + (size_t)j * f;
  float d2 = 0.f;
  for (int t = 0; t < f; ++t) { const float d = xr[t] - yr[t]; d2 += w[t] * d * d; }
  float v = (*sigma2_p) * expf(-0.5f * d2);
  if (jitter != 0.f && i == j) v += jitter;
  out[(size_t)j * n + i] = v;   // write transposed (K_ss is exactly symmetric)
}

__global__ void xi_make(float* xi, const float* m, const float* l,
                        const float* eps, int n) {
  const int i = blockIdx.x * blockDim.x + threadIdx.x;
  if (i < n) xi[i] = expf(0.5f * l[i]) * eps[i] + m[i];
}

// cov_diag[i] = clamp(sigma2 - sum_r Z[r][i]^2, 1e-12)   (Z = inv(L) K_xis^T)
__global__ void colsumsq(const float* __restrict__ Z, int ld, int rows,
                         const float* __restrict__ sigma2_p, float* cov) {
  const int c = blockIdx.x * blockDim.x + threadIdx.x;
  float s = 0.f;
  for (int r = 0; r < rows; ++r) { const float v = Z[(size_t)r * ld + c]; s += v * v; }
  const float cd = *sigma2_p - s;
  cov[c] = cd > 1e-12f ? cd : 1e-12f;
}

__global__ void f_make(float* f, const float* mu, const float* cov,
                       const float* eps, int rows, int cols) {
  const int i = blockIdx.x * blockDim.x + threadIdx.x;
  if (i >= rows * cols) return;
  f[i] = sqrtf(cov[i / cols]) * eps[i] + mu[i];
}

__global__ void z_make(float* z, const float* __restrict__ zm,
                       const float* __restrict__ zch,
                       const float* __restrict__ eps) {
  const size_t idx = (size_t)blockIdx.x * blockDim.x + threadIdx.x;
  if (idx >= (size_t)NB * 1088) return;
  const int b = (int)(idx / 1088), r = (int)(idx % 1088);
  const int t = r / 136, d = r % 136;
  const float* ch = zch + (size_t)b * 36;
  const float* ep = eps + (size_t)b * 1088;
  float acc = zm[idx];
  const int base = t * (t + 1) / 2;
  for (int u = 0; u <= t; ++u) acc += ch[base + u] * ep[u * 136 + d];
  z[idx] = acc;
}

__global__ void concat_xz(float* xz, const float* __restrict__ x,
                          const float* __restrict__ z) {
  const size_t idx = (size_t)blockIdx.x * blockDim.x + threadIdx.x;
  if (idx >= (size_t)NB * 1216) return;
  const int b = (int)(idx / 1216), c = (int)(idx % 1216);
  xz[idx] = (c < 128) ? x[(size_t)b * 128 + c] : z[(size_t)b * 1088 + (c - 128)];
}

// --------------------------------------------------- prior (8x8, one thread)
__global__ void prior_init(float* IK, float* SKH) {
  if (threadIdx.x != 0 || blockIdx.x != 0) return;
  double K[8][8], L[8][8], Li[8][8];
  for (int i = 0; i < 8; ++i)
    for (int j = 0; j < 8; ++j) {
      const double ti = 0.25 * (i + 1), tj = 0.25 * (j + 1);
      K[i][j] = exp(-(ti - tj) * (ti - tj) / 4.0) + (i == j ? 1e-4 : 0.0);
      L[i][j] = 0.0; Li[i][j] = 0.0;
    }
  for (int j = 0; j < 8; ++j) {
    double s = K[j][j];
    for (int p = 0; p < j; ++p) s -= L[j][p] * L[j][p];
    L[j][j] = sqrt(s);
    for (int i = j + 1; i < 8; ++i) {
      double t = K[i][j];
      for (int p = 0; p < j; ++p) t -= L[i][p] * L[j][p];
      L[i][j] = t / L[j][j];
    }
  }
  double skh = 0.0;
  for (int j = 0; j < 8; ++j) skh += log(fabs(L[j][j]));
  for (int c = 0; c < 8; ++c)
    for (int r = c; r < 8; ++r) {
      double s = (r == c) ? 1.0 : 0.0;
      for (int p = c; p < r; ++p) s -= L[r][p] * Li[p][c];
      Li[r][c] = s / L[r][r];
    }
  for (int i = 0; i < 8; ++i)
    for (int j = 0; j < 8; ++j) {
      double s = 0.0;
      for (int p = 0; p < 8; ++p) s += Li[p][i] * Li[p][j];
      IK[i * 8 + j] = (float)s;
    }
  *SKH = (float)skh;
}

// ---------------------------------------------------------------- reductions
__device__ float block_reduce_sum(float v) {
  __shared__ float red[256];
  __syncthreads();
  red[threadIdx.x] = v;
  __syncthreads();
  for (int s = 128; s > 0; s >>= 1) {
    if ((int)threadIdx.x < s) red[threadIdx.x] += red[threadIdx.x + s];
    __syncthreads();
  }
  return red[0];
}

__global__ __launch_bounds__(256) void nll_partial(
    const float* mean, int ldm, int offm, const float* lcov, int ldl, int offl,
    const float* x, int ldx, int rows, int cols, float* part) {
  float local = 0.f;
  const int total = rows * cols;
  for (int idx = blockIdx.x * blockDim.x + threadIdx.x; idx < total;
       idx += gridDim.x * blockDim.x) {
    const int b = idx / cols, c = idx % cols;
    const float m = mean[(size_t)b * ldm + offm + c];
    const float l = lcov[(size_t)b * ldl + offl + c];
    const float d = x[(size_t)b * ldx + c] - m;
    local += l + expf(-l) * d * d + LOG2PI_F;
  }
  const float s = block_reduce_sum(local);
  if (threadIdx.x == 0) part[blockIdx.x] = s;
}

__global__ __launch_bounds__(256) void kldd_partial(
    const float* mu_f, const float* cov, const float* em, const float* el,
    float* part) {
  float local = 0.f;
  const int total = NB * 100;
  for (int idx = blockIdx.x * blockDim.x + threadIdx.x; idx < total;
       idx += gridDim.x * blockDim.x) {
    const int b = idx / 100, c = idx % 100;
    const float mu0 = mu_f[(size_t)b * 100 + c];
    const float ls0 = logf(cov[b]);
    const float mu1 = em[(size_t)b * 120 + 20 + c];
    const float ls1 = el[(size_t)b * 120 + 20 + c];
    const float dm = mu0 - mu1;
    local += ls1 - ls0 - 1.f + expf(ls0 - ls1) + dm * dm * expf(-ls1);
  }
  const float s = block_reduce_sum(local);
  if (threadIdx.x == 0) part[blockIdx.x] = s;
}

__global__ __launch_bounds__(256) void bkdg_partial(
    const float* __restrict__ zm, const float* __restrict__ zch,
    const float* __restrict__ IK, float* part) {
  __shared__ float ik[64];
  __shared__ float svech[136];
  if (threadIdx.x < 64) ik[threadIdx.x] = IK[threadIdx.x];
  for (int d = threadIdx.x; d < 136; d += 256) {
    float v = 0.f;
    for (int t = 0; t < 16; ++t)
      if (d == t * (t + 3) / 2) v = 1.f;
    svech[d] = v;
  }
  __syncthreads();
  float local = 0.f;
  for (int b = blockIdx.x * blockDim.x + threadIdx.x; b < NB;
       b += gridDim.x * blockDim.x) {
    const float* ch = zch + (size_t)b * 36;
    const float* zr = zm + (size_t)b * 1088;
    float Lc[8][8];
    for (int t = 0; t < 8; ++t)
      for (int u = 0; u < 8; ++u)
        Lc[t][u] = (u <= t) ? ch[t * (t + 1) / 2 + u] : 0.f;
    float tr = 0.f;
    for (int t = 0; t < 8; ++t)
      for (int u = 0; u < 8; ++u) {
        float cv = 0.f;
        for (int v = 0; v < 8; ++v) cv += Lc[t][v] * Lc[u][v];
        tr += ik[t * 8 + u] * cv;
      }
    local += 136.f * tr;
    for (int d = 0; d < 136; ++d) {
      float imu[8];
      const float ve = svech[d];
      for (int u = 0; u < 8; ++u) imu[u] = ve - zr[u * 136 + d];
      for (int t = 0; t < 8; ++t) {
        float s = 0.f;
        for (int u = 0; u < 8; ++u) s += ik[t * 8 + u] * imu[u];
        local += s * imu[t];
      }
    }
    float ld = 0.f;
    for (int t = 0; t < 8; ++t) ld += logf(fabsf(ch[t * (t + 3) / 2]));
    local -= 2.f * 136.f * ld;
  }
  const float s = block_reduce_sum(local);
  if (threadIdx.x == 0) part[blockIdx.x] = s;
}

__global__ __launch_bounds__(256) void finalize(const float* part,
                                                const float* SKH, float* out) {
  float s[5];
  for (int seg = 0; seg < 5; ++seg) {
    float local = 0.f;
    for (int i = threadIdx.x; i < NPB; i += 256) local += part[seg * NPB + i];
    s[seg] = block_reduce_sum(local);
  }
  if (threadIdx.x == 0) {
    const float Bf = (float)NB;
    const float nll  = 0.5f * (s[0] - s[1] + s[2]) / Bf;
    const float kldd = 0.5f * s[3] / Bf;
    const float bk = 0.5f * (s[4] - Bf * 1088.f + 2.f * Bf * 136.f * (*SKH)) / Bf;
    out[0] = bk + kldd;
    out[1] = nll;
  }
}

// ------------------------------------------------------------------- driver
static inline void launch_gemm(hipStream_t st, int flags, float* C, int ldc,
                               const float* Cin, int ldcin, const float* A,
                               int lda, const float* B, int ldb,
                               const float* bias, int M, int N, int K) {
  dim3 g((N + 63) / 64, (M + 127) / 128), b(128);
  switch (flags) {
    case GF_TRANSB:
      gemm_wmma_f32<GF_TRANSB><<<g, b, 0, st>>>(C, ldc, Cin, ldcin, A, lda, B, ldb, bias, M, N, K); break;
    case GF_TRANSB | GF_RELU:
      gemm_wmma_f32<GF_TRANSB | GF_RELU><<<g, b, 0, st>>>(C, ldc, Cin, ldcin, A, lda, B, ldb, bias, M, N, K); break;
    case GF_TRANSA:
      gemm_wmma_f32<GF_TRANSA><<<g, b, 0, st>>>(C, ldc, Cin, ldcin, A, lda, B, ldb, bias, M, N, K); break;
    case GF_NEG:
      gemm_wmma_f32<GF_NEG><<<g, b, 0, st>>>(C, ldc, Cin, ldcin, A, lda, B, ldb, bias, M, N, K); break;
    case GF_NEG | GF_TRANSB:
      gemm_wmma_f32<GF_NEG | GF_TRANSB><<<g, b, 0, st>>>(C, ldc, Cin, ldcin, A, lda, B, ldb, bias, M, N, K); break;
    case GF_NEG | GF_TRANSA:
      gemm_wmma_f32<GF_NEG | GF_TRANSA><<<g, b, 0, st>>>(C, ldc, Cin, ldcin, A, lda, B, ldb, bias, M, N, K); break;
    default: break;
  }
}

// workspace layout (float offsets)
#define WS_H1   ((size_t)0)
#define WS_H2   (WS_H1 + 1048576)
#define WS_H3   (WS_H2 + 1048576)
#define WS_H4   (WS_H3 + 1048576)
#define WS_QM   (WS_H4 + 1048576)
#define WS_QL   (WS_QM + 40960)
#define WS_XI   (WS_QL + 40960)
#define WS_KT   (WS_XI + 40960)
#define WS_KSS  (WS_KT + 4194304)
#define WS_MUF  (WS_KSS + 4194304)
#define WS_COV  (WS_MUF + 204800)
#define WS_F    (WS_COV + 2048)
#define WS_Z    (WS_F + 204800)
#define WS_XZ   (WS_Z + 2228224)
#define WS_EM   (WS_XZ + 2490368)
#define WS_EL   (WS_EM + 245760)
#define WS_IK   (WS_EL + 245760)
#define WS_SKH  (WS_IK + 64)
#define WS_PART (WS_SKH + 1)
#define WS_END  (WS_PART + 5 * NPB)

extern "C" void kernel_launch(void* const* d_in, const int* in_sizes, int n_in,
                              void* d_out, int out_size, void* d_ws,
                              size_t ws_size, hipStream_t stream) {
  (void)in_sizes; (void)n_in; (void)out_size;
  if (ws_size < WS_END * sizeof(float)) return;
  const float* x     = (const float*)d_in[0];
  const float* W1    = (const float*)d_in[1];  const float* b1  = (const float*)d_in[2];
  const float* W11   = (const float*)d_in[3];  const float* b11 = (const float*)d_in[4];
  const float* W12   = (const float*)d_in[5];  const float* b12 = (const float*)d_in[6];
  const float* W2    = (const float*)d_in[7];  const float* b2  = (const float*)d_in[8];
  const float* W21   = (const float*)d_in[9];  const float* b21 = (const float*)d_in[10];
  const float* W22   = (const float*)d_in[11]; const float* b22 = (const float*)d_in[12];
  const float* W3    = (const float*)d_in[13]; const float* b3  = (const float*)d_in[14];
  const float* W31   = (const float*)d_in[15]; const float* b31 = (const float*)d_in[16];
  const float* W32   = (const float*)d_in[17]; const float* b32 = (const float*)d_in[18];
  const float* W4    = (const float*)d_in[19]; const float* b4  = (const float*)d_in[20];
  const float* W41   = (const float*)d_in[21]; const float* b41 = (const float*)d_in[22];
  const float* W42   = (const float*)d_in[23]; const float* b42 = (const float*)d_in[24];
  const float* sP    = (const float*)d_in[25];
  const float* tP    = (const float*)d_in[26];
  const float* sig2  = (const float*)d_in[27];
  const float* wP    = (const float*)d_in[28];
  const float* eXi   = (const float*)d_in[29];
  const float* eF    = (const float*)d_in[30];
  const float* eZ    = (const float*)d_in[31];

  float* ws  = (float*)d_ws;
  float* out = (float*)d_out;
  float* H1 = ws + WS_H1; float* H2 = ws + WS_H2; float* H3 = ws + WS_H3;
  float* H4 = ws + WS_H4; float* QM = ws + WS_QM; float* QL = ws + WS_QL;
  float* XI = ws + WS_XI; float* KT = ws + WS_KT; float* KSS = ws + WS_KSS;
  float* MUF = ws + WS_MUF; float* COV = ws + WS_COV; float* F = ws + WS_F;
  float* Z = ws + WS_Z; float* XZ = ws + WS_XZ; float* EM = ws + WS_EM;
  float* EL = ws + WS_EL; float* IK = ws + WS_IK; float* SKH = ws + WS_SKH;
  float* PART = ws + WS_PART;

  float* zm  = out + 2;
  float* zch = zm + (size_t)NB * 1088;
  float* xm  = zch + (size_t)NB * 36;
  float* xl  = xm + (size_t)NB * 128;

  // encoder q(xi|x)
  launch_gemm(stream, GF_TRANSB | GF_RELU, H1, 512, nullptr, 0, x, 128, W1, 128, b1, NB, 512, 128);
  launch_gemm(stream, GF_TRANSB, QM, 20, nullptr, 0, H1, 512, W11, 512, b11, NB, 20, 512);
  launch_gemm(stream, GF_TRANSB, QL, 20, nullptr, 0, H1, 512, W12, 512, b12, NB, 20, 512);
  xi_make<<<(NB * 20 + 255) / 256, 256, 0, stream>>>(XI, QM, QL, eXi, NB * 20);
  prior_init<<<1, 64, 0, stream>>>(IK, SKH);

  // RBF kernel matrices (K_xis stored transposed: KT[j][i] = k(xi_i, s_j))
  const size_t nn = (size_t)NB * NB;
  rbf_kernel<<<(unsigned)((nn + 255) / 256), 256, 0, stream>>>(KT, XI, sP, wP, sig2, NB, 20, 0.f);
  rbf_kernel<<<(unsigned)((nn + 255) / 256), 256, 0, stream>>>(KSS, sP, sP, wP, sig2, NB, 20, 1e-5f);

  // blocked right-looking Cholesky: K_ss = L L^T (lower, in place)
  for (int k = 0; k < NBLK; ++k) {
    chol_diag<<<1, 256, 0, stream>>>(KSS, NB, k * 64);
    const int nb = NBLK - 1 - k;
    if (nb > 0) {
      chol_trsm_panel<<<nb, 64, 0, stream>>>(KSS, NB, k * 64);
      const int r1 = 64 * (k + 1), R = NB - r1;
      float* Ct = KSS + (size_t)r1 * NB + r1;
      const float* P = KSS + (size_t)r1 * NB + 64 * k;
      launch_gemm(stream, GF_NEG | GF_TRANSB, Ct, NB, Ct, NB, P, NB, P, NB, nullptr, R, R, 64);  // SYRK trailing
    }
  }

  // forward solve L Z = K_xis^T (in place on KT)
  for (int i = 0; i < NBLK; ++i) {
    float* Ci = KT + (size_t)(64 * i) * NB;
    if (i > 0)
      launch_gemm(stream, GF_NEG, Ci, NB, Ci, NB, KSS + (size_t)(64 * i) * NB, NB, KT, NB, nullptr, 64, NB, 64 * i);
    trsv_slab<<<8, 256, 0, stream>>>(KSS + (size_t)(64 * i) * NB + 64 * i, NB, Ci, NB, 0);
  }
  // cov_diag = sigma2 - colnorm2(Z)  (exactly diag(K Kss^-1 K^T))
  colsumsq<<<8, 256, 0, stream>>>(KT, NB, NB, sig2, COV);
  // backward solve L^T Y = Z (in place; Y = kk_inv^T)
  for (int i = NBLK - 1; i >= 0; --i) {
    float* Ci = KT + (size_t)(64 * i) * NB;
    if (i < NBLK - 1) {
      const int r1 = 64 * (i + 1), Kr = NB - r1;
      launch_gemm(stream, GF_NEG | GF_TRANSA, Ci, NB, Ci, NB,
                  KSS + (size_t)r1 * NB + 64 * i, NB, KT + (size_t)r1 * NB, NB, nullptr, 64, NB, Kr);
    }
    trsv_slab<<<8, 256, 0, stream>>>(KSS + (size_t)(64 * i) * NB + 64 * i, NB, Ci, NB, 1);
  }

  // mu_f = kk_inv @ t = Y^T @ t
  launch_gemm(stream, GF_TRANSA, MUF, 100, nullptr, 0, KT, NB, tP, 100, nullptr, NB, 100, NB);
  f_make<<<(NB * 100 + 255) / 256, 256, 0, stream>>>(F, MUF, COV, eF, NB, 100);

  // decoder chain
  launch_gemm(stream, GF_TRANSB | GF_RELU, H2, 512, nullptr, 0, F, 100, W2, 100, b2, NB, 512, 100);
  launch_gemm(stream, GF_TRANSB, zm, 1088, nullptr, 0, H2, 512, W21, 512, b21, NB, 1088, 512);
  launch_gemm(stream, GF_TRANSB, zch, 36, nullptr, 0, H2, 512, W22, 512, b22, NB, 36, 512);
  z_make<<<(unsigned)(((size_t)NB * 1088 + 255) / 256), 256, 0, stream>>>(Z, zm, zch, eZ);
  concat_xz<<<(unsigned)(((size_t)NB * 1216 + 255) / 256), 256, 0, stream>>>(XZ, x, Z);
  launch_gemm(stream, GF_TRANSB | GF_RELU, H3, 512, nullptr, 0, XZ, 1216, W3, 1216, b3, NB, 512, 1216);
  launch_gemm(stream, GF_TRANSB, EM, 120, nullptr, 0, H3, 512, W31, 512, b31, NB, 120, 512);
  launch_gemm(stream, GF_TRANSB, EL, 120, nullptr, 0, H3, 512, W32, 512, b32, NB, 120, 512);
  launch_gemm(stream, GF_TRANSB | GF_RELU, H4, 512, nullptr, 0, Z, 1088, W4, 1088, b4, NB, 512, 1088);
  launch_gemm(stream, GF_TRANSB, xm, 128, nullptr, 0, H4, 512, W41, 512, b41, NB, 128, 512);
  launch_gemm(stream, GF_TRANSB, xl, 128, nullptr, 0, H4, 512, W42, 512, b42, NB, 128, 512);

  // losses (deterministic two-stage reductions)
  nll_partial<<<NPB, 256, 0, stream>>>(xm, 128, 0, xl, 128, 0, x, 128, NB, 128, PART + 0 * NPB);
  nll_partial<<<NPB, 256, 0, stream>>>(QM, 20, 0, QL, 20, 0, XI, 20, NB, 20, PART + 1 * NPB);
  nll_partial<<<NPB, 256, 0, stream>>>(EM, 120, 0, EL, 120, 0, XI, 20, NB, 20, PART + 2 * NPB);
  kldd_partial<<<NPB, 256, 0, stream>>>(MUF, COV, EM, EL, PART + 3 * NPB);
  bkdg_partial<<<NPB, 256, 0, stream>>>(zm, zch, IK, PART + 4 * NPB);
  finalize<<<1, 256, 0, stream>>>(PART, SKH, out);
}